// Auto_ODE_SEIR_77532749627470
// MI455X (gfx1250) — compile-verified
//
#include <hip/hip_runtime.h>

// SEIR Euler integration, one thread per batch element.
// Output layout (reference): out[(t*B + b)*4 + c], c in {S,E,I,R}.
// 512 MB streaming writes -> non-temporal b128 stores; compute is negligible.

typedef float v4f __attribute__((ext_vector_type(4)));

__global__ __launch_bounds__(256) void seir_euler_kernel(
    const float* __restrict__ initial,   // [4*B]  rows: S0,E0,I0,R0
    const float* __restrict__ pbeta,     // [1]
    const float* __restrict__ pgamma,    // [1]
    const float* __restrict__ psigma,    // [1]
    const int*   __restrict__ pt,        // [1]  number of timesteps T
    float*       __restrict__ out,       // [T*B*4]
    int B)
{
    const int b = blockIdx.x * blockDim.x + threadIdx.x;
    if (b >= B) return;

    const float step = 0.5f;
    // Uniform scalars -> scalar loads (s_load_b32), broadcast to the wave.
    const float bet = pbeta[0];
    const float gam = pgamma[0];
    const float sig = psigma[0];
    const int   T   = pt[0];

    // State lives in registers for the whole trajectory.
    float S = initial[0 * B + b];
    float E = initial[1 * B + b];
    float I = initial[2 * B + b];
    float R = initial[3 * B + b];

    // Row pointer: advance by B float4's (B*16 bytes) per timestep.
    v4f* __restrict__ p = reinterpret_cast<v4f*>(out) + b;
    const long long rowStride = (long long)B;

    for (int t = 0; t < T; ++t) {
        // Store current state as row t (row 0 == initial state), NT so the
        // 512 MB stream bypasses/rinses L2 (output >> 192 MB L2).
        v4f st = { S, E, I, R };
        __builtin_nontemporal_store(st, p);
        p += rowStride;

        // Euler update; all four updates read the *previous* step's values
        // (matches reference semantics). Shallow dependent chain:
        //   bSI, sE, gI are computable in parallel, then 4 independent FMAs.
        const float bSI = bet * S * I;
        const float sE  = sig * E;
        const float gI  = gam * I;
        const float Sn = S - bSI * step;
        const float En = E + (bSI - sE) * step;
        const float In = I + (sE - gI) * step;
        const float Rn = R + gI * step;
        S = Sn; E = En; I = In; R = Rn;
    }
}

extern "C" void kernel_launch(void* const* d_in, const int* in_sizes, int n_in,
                              void* d_out, int out_size, void* d_ws, size_t ws_size,
                              hipStream_t stream) {
    // setup_inputs() order: initial (4*B f32), beta (1 f32), gamma (1 f32),
    // sigma (1 f32), t (1 int).
    const float* initial = (const float*)d_in[0];
    const float* beta    = (const float*)d_in[1];
    const float* gamma   = (const float*)d_in[2];
    const float* sigma   = (const float*)d_in[3];
    const int*   t       = (const int*)d_in[4];

    const int B = in_sizes[0] / 4;

    const int threads = 256;                      // 8 wave32s per block
    const int blocks  = (B + threads - 1) / threads;

    seir_euler_kernel<<<blocks, threads, 0, stream>>>(
        initial, beta, gamma, sigma, t, (float*)d_out, B);
}